// SudokuGNN_27986006900903
// MI455X (gfx1250) — compile-verified
//
#include <hip/hip_runtime.h>

#define N_NODES 165888
#define N_EDGES 3317760
#define HID 64
#define N_LAYERS 4
#define LN_EPS 1e-5f

// K-pair interleaved weight layout: pair row stride in floats.
// 160 mod 64 == 32 -> the two wave halves (pair rows p0, p0+1) hit disjoint
// LDS bank sets, so the ds_load_b64 B-fragment reads are conflict-free.
#define WSTRIDE 160
#define WMAT (32 * WSTRIDE)  // floats per staged weight matrix

typedef __attribute__((ext_vector_type(2))) float v2f;
typedef __attribute__((ext_vector_type(8))) float v8f;

__device__ __forceinline__ v8f wmma_f32(v2f a, v2f b, v8f c) {
  // (neg_a, A, neg_b, B, c_mod, C, reuse_a, reuse_b) -> v_wmma_f32_16x16x4_f32
  return __builtin_amdgcn_wmma_f32_16x16x4_f32(false, a, false, b, (short)0, c,
                                               false, false);
}

// Stage W[64][64] (used as W^T) into LDS, K-pair interleaved:
//   wt[p*WSTRIDE + n*2 + j] = W[n][2p + j],  p = 0..31, j = 0..1
// so B-fragment (K=2p+kb..+1, N=n) is one contiguous 8B LDS load.
__device__ __forceinline__ void stage_weights(float* wt,
                                              const float* __restrict__ W,
                                              int tid) {
  for (int idx = tid; idx < 4096; idx += 256) {
    int n = idx >> 6, k = idx & 63;  // coalesced global read
    wt[(k >> 1) * WSTRIDE + n * 2 + (k & 1)] = W[idx];
  }
}

// ---------------- input projection: h = x @ W_in^T + b_in ----------------
__global__ void input_proj_kernel(const float* __restrict__ x,
                                  const float* __restrict__ W_in,
                                  const float* __restrict__ b_in,
                                  float* __restrict__ h) {
  __shared__ float w[HID * 9];
  __shared__ float bs[HID];
  int tid = threadIdx.x;
  for (int i = tid; i < HID * 9; i += 256) w[i] = W_in[i];
  if (tid < HID) bs[tid] = b_in[tid];
  __syncthreads();
  int n = blockIdx.x * 4 + (tid >> 6);
  int j = tid & 63;
  const float* xr = x + (size_t)n * 9;
  float acc = bs[j];
#pragma unroll
  for (int k = 0; k < 9; ++k) acc += xr[k] * w[j * 9 + k];
  h[(size_t)n * HID + j] = acc;
}

// ------------- residual: out = A @ W^T + bias (fp32 WMMA) ----------------
__global__ void gemm_bias_kernel(const float* __restrict__ A,
                                 const float* __restrict__ W,
                                 const float* __restrict__ bias,
                                 float* __restrict__ out) {
  __shared__ float wt[WMAT];
  int tid = threadIdx.x;
  stage_weights(wt, W, tid);
  __syncthreads();

  int wave = tid >> 5, lane = tid & 31;
  int m0 = blockIdx.x * 128 + wave * 16;
  int m = lane & 15;
  int half = lane >> 4;        // 0: K sub-offset 0, 1: K sub-offset 2
  int nlo = lane & 15;
  const float* arow = A + (size_t)(m0 + m) * HID;

  v8f acc[4] = {};
  for (int k = 0; k < 64; k += 4) {
    v2f a = *(const v2f*)(arow + k + half * 2);
    const float* wp = wt + ((k >> 1) + half) * WSTRIDE;
#pragma unroll
    for (int t = 0; t < 4; ++t) {
      v2f b = *(const v2f*)(wp + (t * 16 + nlo) * 2);  // one ds_load_b64
      acc[t] = wmma_f32(a, b, acc[t]);
    }
  }
#pragma unroll
  for (int t = 0; t < 4; ++t) {
    int col = t * 16 + nlo;
    float bv = bias[col];
#pragma unroll
    for (int r = 0; r < 8; ++r) {
      int mm = r + (half << 3);
      out[(size_t)(m0 + mm) * HID + col] = acc[t][r] + bv;
    }
  }
}

// ---------------------------- zero agg -----------------------------------
__global__ void zero_kernel(float4* __restrict__ p, int n4) {
  int i = blockIdx.x * 256 + threadIdx.x;
  if (i < n4) p[i] = make_float4(0.f, 0.f, 0.f, 0.f);
}

// ------------------- scatter-add: agg[dst] += h[src] ---------------------
__global__ void scatter_add_kernel(const float* __restrict__ h,
                                   const int* __restrict__ ei,
                                   float* __restrict__ agg) {
  long long t = (long long)blockIdx.x * 256 + threadIdx.x;
  int e = (int)(t >> 4);
  int c4 = ((int)t & 15) * 4;
  if (e >= N_EDGES) return;
  int src = ei[e];
  int dst = ei[N_EDGES + e];
  const float4 v = *(const float4*)(h + (size_t)src * HID + c4);
  float* ap = agg + (size_t)dst * HID + c4;
  atomicAdd(ap + 0, v.x);
  atomicAdd(ap + 1, v.y);
  atomicAdd(ap + 2, v.z);
  atomicAdd(ap + 3, v.w);
}

// - fused conv layer: h = LN(h@Wroot^T + agg@Wrel^T + b)·leaky + residual -
__global__ void conv_layer_kernel(float* __restrict__ h,  // in/out [N,64]
                                  const float* __restrict__ agg,
                                  const float* __restrict__ W_root,
                                  const float* __restrict__ W_rel,
                                  const float* __restrict__ b_conv,
                                  const float* __restrict__ resid,
                                  const float* __restrict__ ln_g,
                                  const float* __restrict__ ln_b) {
  // Phase A: two staged weight matrices (2*WMAT floats = 40 KB).
  // Phase B: 128x65 output tile (8320 floats) -- aliased, barrier-separated.
  __shared__ float smem[2 * WMAT];
  float* wroot = smem;
  float* wrel = smem + WMAT;
  int tid = threadIdx.x;
  stage_weights(wroot, W_root, tid);
  stage_weights(wrel, W_rel, tid);
  __syncthreads();

  int wave = tid >> 5, lane = tid & 31;
  int m0 = blockIdx.x * 128 + wave * 16;
  int m = lane & 15;
  int half = lane >> 4;
  int nlo = lane & 15;
  const float* hrow = h + (size_t)(m0 + m) * HID;
  const float* grow = agg + (size_t)(m0 + m) * HID;

  v8f acc[4] = {};
  for (int k = 0; k < 64; k += 4) {
    v2f ah = *(const v2f*)(hrow + k + half * 2);
    v2f ag = *(const v2f*)(grow + k + half * 2);
    const float* wrp = wroot + ((k >> 1) + half) * WSTRIDE;
    const float* wlp = wrel + ((k >> 1) + half) * WSTRIDE;
#pragma unroll
    for (int t = 0; t < 4; ++t) {
      v2f br = *(const v2f*)(wrp + (t * 16 + nlo) * 2);
      acc[t] = wmma_f32(ah, br, acc[t]);
      v2f bl = *(const v2f*)(wlp + (t * 16 + nlo) * 2);
      acc[t] = wmma_f32(ag, bl, acc[t]);
    }
  }
  __syncthreads();  // all waves done reading weights; smem becomes tile

  float* tile = smem;  // [128][65]
#pragma unroll
  for (int t = 0; t < 4; ++t) {
#pragma unroll
    for (int r = 0; r < 8; ++r) {
      int mm = r + (half << 3);
      tile[(wave * 16 + mm) * 65 + t * 16 + nlo] = acc[t][r];
    }
  }
  __syncthreads();

  // Epilogue: one thread per row (bias + LayerNorm + leaky + residual).
  if (tid < 128) {
    const float* tr = tile + tid * 65;
    float mu = 0.f;
#pragma unroll 8
    for (int j = 0; j < 64; ++j) mu += tr[j] + b_conv[j];
    mu *= (1.f / 64.f);
    float var = 0.f;
#pragma unroll 8
    for (int j = 0; j < 64; ++j) {
      float d = tr[j] + b_conv[j] - mu;
      var += d * d;
    }
    var *= (1.f / 64.f);
    float rs = rsqrtf(var + LN_EPS);
    size_t row = (size_t)blockIdx.x * 128 + tid;
    const float* rr = resid + row * HID;
    float* ho = h + row * HID;
#pragma unroll 8
    for (int j = 0; j < 64; ++j) {
      float y = (tr[j] + b_conv[j] - mu) * rs * ln_g[j] + ln_b[j];
      y = y > 0.f ? y : 0.01f * y;
      ho[j] = y + rr[j];
    }
  }
}

// ---------------- output projection: out = h @ W_out^T + b_out -----------
__global__ void out_proj_kernel(const float* __restrict__ h,
                                const float* __restrict__ W_out,
                                const float* __restrict__ b_out,
                                float* __restrict__ out) {
  __shared__ float w[9 * 64];
  __shared__ float bo[9];
  int tid = threadIdx.x;
  for (int i = tid; i < 9 * 64; i += 256) w[i] = W_out[i];
  if (tid < 9) bo[tid] = b_out[tid];
  __syncthreads();
  int n = blockIdx.x * 256 + tid;
  if (n >= N_NODES) return;
  const float* hr = h + (size_t)n * HID;
  float acc[9];
#pragma unroll
  for (int j = 0; j < 9; ++j) acc[j] = bo[j];
  for (int k = 0; k < 64; ++k) {
    float hv = hr[k];
#pragma unroll
    for (int j = 0; j < 9; ++j) acc[j] += hv * w[j * 64 + k];
  }
#pragma unroll
  for (int j = 0; j < 9; ++j) out[(size_t)n * 9 + j] = acc[j];
}

extern "C" void kernel_launch(void* const* d_in, const int* in_sizes, int n_in,
                              void* d_out, int out_size, void* d_ws,
                              size_t ws_size, hipStream_t stream) {
  (void)in_sizes; (void)n_in; (void)out_size; (void)ws_size;
  const float* x      = (const float*)d_in[0];
  const int*   ei     = (const int*)d_in[1];
  const float* W_in   = (const float*)d_in[2];
  const float* b_in   = (const float*)d_in[3];
  const float* W_rel  = (const float*)d_in[4];   // [4,64,64]
  const float* W_root = (const float*)d_in[5];   // [4,64,64]
  const float* b_conv = (const float*)d_in[6];   // [4,64]
  const float* W_res  = (const float*)d_in[7];
  const float* b_res  = (const float*)d_in[8];
  const float* ln_g   = (const float*)d_in[9];
  const float* ln_b   = (const float*)d_in[10];
  const float* W_out  = (const float*)d_in[11];
  const float* b_out  = (const float*)d_in[12];
  float* out = (float*)d_out;

  size_t NH = (size_t)N_NODES * HID;
  float* h     = (float*)d_ws;
  float* agg   = h + NH;
  float* resid = agg + NH;

  input_proj_kernel<<<N_NODES / 4, 256, 0, stream>>>(x, W_in, b_in, h);
  gemm_bias_kernel<<<N_NODES / 128, 256, 0, stream>>>(h, W_res, b_res, resid);

  for (int l = 0; l < N_LAYERS; ++l) {
    zero_kernel<<<(int)((NH / 4 + 255) / 256), 256, 0, stream>>>((float4*)agg,
                                                                 (int)(NH / 4));
    scatter_add_kernel<<<(int)(((long long)N_EDGES * 16) / 256), 256, 0,
                         stream>>>(h, ei, agg);
    conv_layer_kernel<<<N_NODES / 128, 256, 0, stream>>>(
        h, agg, W_root + (size_t)l * HID * HID, W_rel + (size_t)l * HID * HID,
        b_conv + (size_t)l * HID, resid, ln_g, ln_b);
  }

  out_proj_kernel<<<(N_NODES + 255) / 256, 256, 0, stream>>>(h, W_out, b_out,
                                                             out);
}